// GCNlayer_48215302864915
// MI455X (gfx1250) — compile-verified
//
#include <hip/hip_runtime.h>
#include <hip/hip_bf16.h>

// GCN layer: Z = segment_sum(X[src]*w, dst) @ W + b
// Reordered as Z = segment_sum(Y[src]*w, dst) + b with Y = X @ W (exact same
// linear operator; fp32 throughout, matching reference precision).
//
// N=50000 nodes, E=800000 edges, D=128.

#define GCN_N 50000
#define GCN_E 800000
#define GCN_D 128

typedef __attribute__((ext_vector_type(2))) float v2f;
typedef __attribute__((ext_vector_type(8))) float v8f;

// ---------------------------------------------------------------------------
// Kernel 1: Y = X @ W  using V_WMMA_F32_16X16X4_F32 (wave32, fp32 matrix pipe).
// One 16x16 tile of Y per wave; K=128 -> 32 WMMA ops, fully unrolled.
// Grid: 3125 blocks x 256 threads = 25000 waves = 3125 Mtiles * 8 Ntiles exactly.
// ---------------------------------------------------------------------------
__global__ __launch_bounds__(256) void gcn_xw_wmma(const float* __restrict__ X,
                                                   const float* __restrict__ W,
                                                   float* __restrict__ Y) {
    const int lane = threadIdx.x & 31;
    const int waveId = blockIdx.x * 8 + (threadIdx.x >> 5);
    const int mt = waveId >> 3;   // 0..3124
    const int nt = waveId & 7;    // 0..7
    const int m0 = mt * 16;
    const int n0 = nt * 16;
    const int r    = lane & 15;   // row (A) / col (B,C,D) within tile
    const int half = lane >> 4;   // lane group selects K pair / M offset

    v8f acc = {};  // 16x16 f32 C/D: VGPR i holds row (i + 8*half), col r

    const float* __restrict__ arow = X + (size_t)(m0 + r) * GCN_D;

#pragma unroll
    for (int k = 0; k < GCN_D; k += 4) {
        const int ka = k + 2 * half;           // this lane-group's K pair
        v2f a;                                  // A 16x4: lane r, K=ka..ka+1
        a.x = arow[ka];
        a.y = arow[ka + 1];
        v2f bv;                                 // B 4x16: col r, K=ka..ka+1
        bv.x = W[(size_t)ka * GCN_D + n0 + r];
        bv.y = W[(size_t)(ka + 1) * GCN_D + n0 + r];
        // 8 args: (neg_a, A, neg_b, B, c_mod, C, reuse_a, reuse_b)
        acc = __builtin_amdgcn_wmma_f32_16x16x4_f32(
            false, a, false, bv, (short)0, acc, false, false);
    }

#pragma unroll
    for (int i = 0; i < 8; ++i) {
        const int row = m0 + i + 8 * half;
        Y[(size_t)row * GCN_D + n0 + r] = acc[i];
    }
}

// ---------------------------------------------------------------------------
// Kernel 2: Z[n, f] = b[f]  (deterministic re-init of d_out each call)
// ---------------------------------------------------------------------------
__global__ __launch_bounds__(256) void gcn_bias_init(const float* __restrict__ b,
                                                     float* __restrict__ Z) {
    const size_t i = (size_t)blockIdx.x * 256 + threadIdx.x;
    if (i < (size_t)GCN_N * GCN_D) {
        Z[i] = b[i & (GCN_D - 1)];
    }
}

// ---------------------------------------------------------------------------
// Kernel 3: edge scatter — one wave per edge. Lane l covers features
// [4l, 4l+4): float4 load from Y[src], scale by w, 4 native f32 atomics into
// Z[dst]. Y and Z are both L2-resident (25.6 MB each vs 192 MB L2), so this
// runs at L2 atomic throughput.
// ---------------------------------------------------------------------------
__global__ __launch_bounds__(256) void gcn_edge_scatter(const float* __restrict__ Y,
                                                        const int* __restrict__ esrc,
                                                        const int* __restrict__ edst,
                                                        const float* __restrict__ ew,
                                                        float* __restrict__ Z) {
    const int lane = threadIdx.x & 31;
    const int e = blockIdx.x * 8 + (threadIdx.x >> 5);
    if (e >= GCN_E) return;

    const int s = esrc[e];
    const int d = edst[e];
    const float w = ew[e];

    const float4 y = *(const float4*)(Y + (size_t)s * GCN_D + lane * 4);
    float* zp = Z + (size_t)d * GCN_D + lane * 4;

    unsafeAtomicAdd(zp + 0, w * y.x);   // global_atomic_add_f32
    unsafeAtomicAdd(zp + 1, w * y.y);
    unsafeAtomicAdd(zp + 2, w * y.z);
    unsafeAtomicAdd(zp + 3, w * y.w);
}

// ---------------------------------------------------------------------------
extern "C" void kernel_launch(void* const* d_in, const int* in_sizes, int n_in,
                              void* d_out, int out_size, void* d_ws, size_t ws_size,
                              hipStream_t stream) {
    const float* X    = (const float*)d_in[0];
    const int*   esrc = (const int*)  d_in[1];
    const int*   edst = (const int*)  d_in[2];
    const float* ew   = (const float*)d_in[3];
    const float* W    = (const float*)d_in[4];
    const float* b    = (const float*)d_in[5];
    float* Z = (float*)d_out;
    float* Y = (float*)d_ws;   // 50000*128*4 = 25.6 MB scratch

    // 1) Y = X @ W  (WMMA fp32)
    gcn_xw_wmma<<<3125, 256, 0, stream>>>(X, W, Y);

    // 2) Z = broadcast(b)
    gcn_bias_init<<<(GCN_N * GCN_D + 255) / 256, 256, 0, stream>>>(b, Z);

    // 3) Z[dst] += w * Y[src]  (one wave per edge)
    gcn_edge_scatter<<<(GCN_E + 7) / 8, 256, 0, stream>>>(Y, esrc, edst, ew, Z);
}